// SSIMLoss_13718125543429
// MI455X (gfx1250) — compile-verified
//
#include <hip/hip_runtime.h>

// ---------------------------------------------------------------------------
// SSIM loss on MI455X (gfx1250): separable 11x11 Gaussian depthwise conv
// mapped onto V_WMMA_F32_16X16X4_F32 as banded matmuls, fused with the
// pointwise SSIM map and a deterministic two-stage reduction.
//
// WMMA layout assumptions (CDNA5 ISA 7.12.2, f32 16x16x4):
//   A (16x4): lanes 0-15 -> M=lane, VGPR0=K0,VGPR1=K1 ; lanes16-31 -> K2,K3
//   B (4x16): VGPR0: lanes0-15 K=0 / lanes16-31 K=2 ; VGPR1: K=1 / K=3 (N=lane%16)
//   C/D (16x16): VGPR v: lanes0-15 M=v, N=lane ; lanes16-31 M=v+8
//
// Wave-uniform subtile loops are driven through readfirstlane so control flow
// is scalar (EXEC stays all-ones at every WMMA, as the ISA requires).
// ---------------------------------------------------------------------------

typedef __attribute__((ext_vector_type(2))) float v2f;
typedef __attribute__((ext_vector_type(8))) float v8f;

#define NBLOCKS 3072
#define TILES_PER_BLOCK 16      // 49152 tiles total = 96 * 32 * 16

#define TW 32                   // tile width (output)
#define THT 16                  // tile height (output)
#define IN_ROWS 26              // THT + 10 halo
#define IN_STRIDE 44            // 42 data cols + 2 zero pad cols
#define PLANE_SZ (IN_ROWS * IN_STRIDE)   // 1144
#define H_STRIDE 33
#define H_ROWS 32
#define H_SZ (H_ROWS * H_STRIDE)         // 1056
#define S_STRIDE 33
#define S_SZ (THT * S_STRIDE)            // 528

// normalized gaussian window, sigma=1.5, ws=11
__device__ __constant__ float GW[11] = {
    0.00102838f, 0.00759876f, 0.03600078f, 0.10936070f, 0.21300554f,
    0.26601173f, 0.21300554f, 0.10936070f, 0.03600078f, 0.00759876f,
    0.00102838f};

__device__ __forceinline__ float band(int d) {
  return (d >= 0 && d <= 10) ? GW[d] : 0.0f;
}

__global__ __launch_bounds__(256) void ssim_tile_kernel(
    const float* __restrict__ img_gen, const float* __restrict__ img_ref,
    float* __restrict__ partial) {
  __shared__ float lds[5 * PLANE_SZ + 5 * H_SZ];   // 44 KB
  float* planes = lds;
  float* Hbuf = lds + 5 * PLANE_SZ;
  float* stats = lds;  // aliases planes (dead after horizontal pass + barrier)
  __shared__ float red[256];

  const int tid = threadIdx.x;
  // wave index as an SGPR -> scalar loop control around the WMMA loops
  const int wave = __builtin_amdgcn_readfirstlane(tid >> 5);
  const int lane = tid & 31;
  const int half = lane >> 4;   // 0: lanes 0-15, 1: lanes 16-31
  const int ln = lane & 15;

  // Banded coefficient fragments: identical expression serves as the B
  // operand of the horizontal pass and the A operand of the vertical pass.
  v2f coef[7];
#pragma unroll
  for (int st = 0; st < 7; ++st) {
    int k0 = 4 * st + 2 * half;
    coef[st].x = band(k0 - ln);
    coef[st].y = band(k0 + 1 - ln);
  }

  float acc = 0.0f;

  for (int t = 0; t < TILES_PER_BLOCK; ++t) {
    const int tile = blockIdx.x + t * NBLOCKS;
    const int tx = tile & 15;          // 16 col tiles of 32
    const int ty = (tile >> 4) & 31;   // 32 row tiles of 16
    const int nc = tile >> 9;          // batch*channel plane (0..95)
    const float* gbase = img_gen + (size_t)nc * 262144u;
    const float* rbase = img_ref + (size_t)nc * 262144u;
    const int gx0 = tx * TW;
    const int gy0 = ty * THT;

    // ---- phase 1: build the 5 stat planes in LDS (zero "same" padding) ----
    for (int idx = tid; idx < IN_ROWS * IN_STRIDE; idx += 256) {
      int r = idx / IN_STRIDE;
      int c = idx - r * IN_STRIDE;
      float g = 0.0f, q = 0.0f;
      if (c < 42) {
        int gy = gy0 - 5 + r;
        int gx = gx0 - 5 + c;
        if (gy >= 0 && gy < 512 && gx >= 0 && gx < 512) {
          int o = gy * 512 + gx;
          g = (gbase[o] + 1.0f) * 0.5f;
          q = (rbase[o] + 1.0f) * 0.5f;
        }
      }
      planes[0 * PLANE_SZ + idx] = g;
      planes[1 * PLANE_SZ + idx] = q;
      planes[2 * PLANE_SZ + idx] = g * g;
      planes[3 * PLANE_SZ + idx] = q * q;
      planes[4 * PLANE_SZ + idx] = g * q;
    }

    // ---- prefetch next tile's halo region while we compute this one ----
    if (t + 1 < TILES_PER_BLOCK) {
      const int tile_n = blockIdx.x + (t + 1) * NBLOCKS;
      const int txn = tile_n & 15;
      const int tyn = (tile_n >> 4) & 31;
      const int ncn = tile_n >> 9;
      const float* gb = img_gen + (size_t)ncn * 262144u;
      const float* rb = img_ref + (size_t)ncn * 262144u;
      // 26 rows x (2 cacheline probes) x (2 images) = 104 prefetches
      for (int idx = tid; idx < IN_ROWS * 4; idx += 256) {
        int r = idx >> 2;
        int seg = idx & 3;               // bit0: image, bit1: line half
        int gy = tyn * THT - 5 + r;
        int gx = txn * TW - 5 + ((seg >> 1) << 5);
        if (gy < 0) gy = 0; else if (gy > 511) gy = 511;
        if (gx < 0) gx = 0; else if (gx > 511) gx = 511;
        const float* p = ((seg & 1) ? rb : gb) + gy * 512 + gx;
        __builtin_prefetch(p, 0, 0);     // -> global_prefetch_b8
      }
    }
    __syncthreads();

    // ---- phase 2: horizontal pass, A=data(16x4), B=band coefs -> Hbuf ----
    // 5 planes x 2 row-blocks x 2 col-blocks = 20 subtiles across 8 waves
    for (int s = wave; s < 20; s += 8) {
      int p = s >> 2, r4 = s & 3;
      int rb = r4 >> 1, cb = r4 & 1;
      int row = rb * 16 + ln;
      if (row > 25) row = 25;  // clamp: results for rows>25 get zero coefs later
      const float* prow = planes + p * PLANE_SZ + row * IN_STRIDE + cb * 16;
      v8f c = {};
#pragma unroll
      for (int st = 0; st < 7; ++st) {
        int k0 = 4 * st + 2 * half;
        v2f a;
        a.x = prow[k0];
        a.y = prow[k0 + 1];
        c = __builtin_amdgcn_wmma_f32_16x16x4_f32(false, a, false, coef[st],
                                                  (short)0, c, false, false);
      }
      float* hb = Hbuf + p * H_SZ + (rb * 16 + 8 * half) * H_STRIDE + cb * 16 + ln;
#pragma unroll
      for (int v = 0; v < 8; ++v) hb[v * H_STRIDE] = c[v];
    }
    __syncthreads();

    // ---- phase 3: vertical pass, A=band coefs, B=data(4x16) -> stats ----
    // 5 planes x 2 col-blocks = 10 subtiles across 8 waves
    for (int s = wave; s < 10; s += 8) {
      int p = s >> 1, cb = s & 1;
      const float* hc = Hbuf + p * H_SZ + cb * 16 + ln;
      v8f c = {};
#pragma unroll
      for (int st = 0; st < 7; ++st) {
        int k0 = 4 * st + 2 * half;
        v2f b;
        b.x = hc[k0 * H_STRIDE];
        b.y = hc[(k0 + 1) * H_STRIDE];
        c = __builtin_amdgcn_wmma_f32_16x16x4_f32(false, coef[st], false, b,
                                                  (short)0, c, false, false);
      }
      float* sb = stats + p * S_SZ + (8 * half) * S_STRIDE + cb * 16 + ln;
#pragma unroll
      for (int v = 0; v < 8; ++v) sb[v * S_STRIDE] = c[v];
    }
    __syncthreads();

    // ---- phase 4: pointwise SSIM map + local accumulation ----
    for (int idx = tid; idx < THT * TW; idx += 256) {
      int y = idx >> 5, x = idx & 31;
      int o = y * S_STRIDE + x;
      float mu1 = stats[0 * S_SZ + o];
      float mu2 = stats[1 * S_SZ + o];
      float egg = stats[2 * S_SZ + o];
      float err = stats[3 * S_SZ + o];
      float egr = stats[4 * S_SZ + o];
      float mu1sq = mu1 * mu1;
      float mu2sq = mu2 * mu2;
      float mu12 = mu1 * mu2;
      float s1 = egg - mu1sq;
      float s2 = err - mu2sq;
      float s12 = egr - mu12;
      float num = (2.0f * mu12 + 6.5025f) * (2.0f * s12 + 58.5225f);
      float den = (mu1sq + mu2sq + 6.5025f) * (s1 + s2 + 58.5225f);
      acc += num / den;
    }
    __syncthreads();  // stats (== planes) may be overwritten next iteration
  }

  // ---- deterministic block reduction -> one partial per block ----
  red[tid] = acc;
  __syncthreads();
  for (int off = 128; off > 0; off >>= 1) {
    if (tid < off) red[tid] += red[tid + off];
    __syncthreads();
  }
  if (tid == 0) partial[blockIdx.x] = red[0];
}

__global__ __launch_bounds__(256) void ssim_final_kernel(
    const float* __restrict__ partial, float* __restrict__ out) {
  __shared__ float red[256];
  float s = 0.0f;
  for (int i = threadIdx.x; i < NBLOCKS; i += 256) s += partial[i];
  red[threadIdx.x] = s;
  __syncthreads();
  for (int off = 128; off > 0; off >>= 1) {
    if (threadIdx.x < off) red[threadIdx.x] += red[threadIdx.x + off];
    __syncthreads();
  }
  if (threadIdx.x == 0) out[0] = 1.0f - red[0] / 25165824.0f;
}

extern "C" void kernel_launch(void* const* d_in, const int* in_sizes, int n_in,
                              void* d_out, int out_size, void* d_ws,
                              size_t ws_size, hipStream_t stream) {
  const float* img_gen = (const float*)d_in[0];
  const float* img_ref = (const float*)d_in[1];
  float* partial = (float*)d_ws;  // 3072 floats of scratch
  ssim_tile_kernel<<<NBLOCKS, 256, 0, stream>>>(img_gen, img_ref, partial);
  ssim_final_kernel<<<1, 256, 0, stream>>>(partial, (float*)d_out);
}